// MeanAggregator_47777216201141
// MI455X (gfx1250) — compile-verified
//
#include <hip/hip_runtime.h>
#include <stdint.h>

// Static shapes from the reference (input_specs are fixed).
#define H_DIM 256
#define T_DIM 32
#define SEQ_LEN 10
#define OUT_DIM (H_DIM + T_DIM)      // 288
#define NODES_PER_SEG 10
#define PAD_TIME 1000000.0f
#define WAVES_PER_BLOCK 4

typedef float v4f __attribute__((ext_vector_type(4)));
typedef unsigned int v4u __attribute__((ext_vector_type(4)));
typedef int v4i __attribute__((ext_vector_type(4)));
typedef int v8i __attribute__((ext_vector_type(8)));

// ---------------------------------------------------------------------------
// Kernel A: map[slot] = -1 for all slots.
// ---------------------------------------------------------------------------
__global__ __launch_bounds__(256)
void map_init_kernel(int* __restrict__ map, int n_slots) {
  int i = blockIdx.x * blockDim.x + threadIdx.x;
  if (i < n_slots) map[i] = -1;
}

// ---------------------------------------------------------------------------
// Kernel B: invert the (seg_example, seg_pos) scatter: map[ex*SEQ_LEN+pos]=seg.
// ---------------------------------------------------------------------------
__global__ __launch_bounds__(256)
void map_scatter_kernel(int* __restrict__ map,
                        const int* __restrict__ seg_example,
                        const int* __restrict__ seg_pos,
                        int n_seg) {
  int i = blockIdx.x * blockDim.x + threadIdx.x;
  if (i < n_seg) map[seg_example[i] * SEQ_LEN + seg_pos[i]] = i;
}

// ---------------------------------------------------------------------------
// Kernel C: wave-per-slot; every output row written exactly once.
//   Filled slot : the 10 neighbor rows (1 KB each) are gathered into LDS by
//                 the Tensor Data Mover in GATHER mode: D# with gather_mode=1,
//                 32-bit row indices (8 per descriptor), tile_dim0=256 x 4B.
//                 Two TENSOR_LOAD_TO_LDS ops (inline asm; ISA operand order
//                 VADDR0=4 SGPRs, VADDR1=8 SGPRs, VADDR2/3=4 SGPRs) move all
//                 10 rows; wave waits on s_wait_tensorcnt 0, then lane-reduces.
//   Pad slot    : embed dims = 0, t = 1e6.
//   Both        : time dims = cos(t * t_w + t_b), one lane per dim (T_DIM=32).
//   All stores non-temporal (output is write-once, no reuse).
// ---------------------------------------------------------------------------
__global__ __launch_bounds__(WAVES_PER_BLOCK * 32)
void out_kernel(const float* __restrict__ ent,
                const int* __restrict__ flat_s,
                const int* __restrict__ time_vals,
                const float* __restrict__ t_w,
                const float* __restrict__ t_b,
                const int* __restrict__ map,
                float* __restrict__ out,
                int n_slots, unsigned n_ent, float inv_cnt) {
  // 10 KB staging per wave, 40 KB per block (<=8 blocks per 320 KB WGP-LDS).
  __shared__ __align__(16) float stage[WAVES_PER_BLOCK][NODES_PER_SEG * H_DIM];

  const int lane = threadIdx.x & 31;
  const int wave = threadIdx.x >> 5;
  const int slot = blockIdx.x * WAVES_PER_BLOCK + wave;
  if (slot >= n_slots) return;               // uniform per wave

  const int seg = map[slot];                 // uniform per wave
  float* orow = out + (long)slot * OUT_DIM;
  float t;

  if (seg >= 0) {
    // ---- TDM gather + mean path ----
    const int segU = __builtin_amdgcn_readfirstlane(seg);

    // 10 wave-uniform row indices -> SGPRs.
    unsigned idx[NODES_PER_SEG];
    #pragma unroll
    for (int k = 0; k < NODES_PER_SEG; ++k)
      idx[k] = (unsigned)__builtin_amdgcn_readfirstlane(
                   flat_s[segU * NODES_PER_SEG + k]);

    // Staging LDS address is wave-uniform; force SGPR-provable uniformity.
    const unsigned ldsA = (unsigned)__builtin_amdgcn_readfirstlane(
        (int)(unsigned)(uintptr_t)(&stage[wave][0]));
    const uint64_t ea = (uint64_t)(uintptr_t)ent;

    // D# group 0: count=1 | gather_index_size(32b)<<30 | gather_mode<<31,
    //             lds_addr, global_addr[31:0], global_addr[56:32] | type=2.
    v4u g0 = { 1u | (1u << 30) | (1u << 31),
               ldsA,
               (unsigned)ea,
               ((unsigned)(ea >> 32) & 0x01FFFFFFu) | (2u << 30) };

    // D# group 1: data_size=4B; tensor_dim0=H_DIM; tensor_dim1=n_ent;
    //             tile_dim0=H_DIM; tile_dim1=#indices; dim0_stride=H_DIM.
    v8i g1 = { (int)(2u << 16),                                  // data_size
               (int)((unsigned)H_DIM << 16),                     // dim0 lo16
               (int)((n_ent & 0xFFFFu) << 16),                   // dim1 lo16
               (int)((n_ent >> 16) | ((unsigned)H_DIM << 16)),   // dim1 hi|tile0
               8,                                                // tile_dim1
               H_DIM,                                            // dim0 stride
               0, 0 };

    // Descriptor 1: rows 0..7.
    v4i g2 = { (int)idx[0], (int)idx[1], (int)idx[2], (int)idx[3] };
    v4i g3 = { (int)idx[4], (int)idx[5], (int)idx[6], (int)idx[7] };
    asm volatile("tensor_load_to_lds %0, %1, %2, %3"
                 :: "s"(g0), "s"(g1), "s"(g2), "s"(g3)
                 : "memory");

    // Descriptor 2: rows 8..9 land right after the first 8 KB.
    g0.y = ldsA + 8u * (unsigned)(H_DIM * 4);
    g1[4] = 2;                                                   // 2 indices
    v4i h2 = { (int)idx[8], (int)idx[9], 0, 0 };
    v4i h3 = { 0, 0, 0, 0 };
    asm volatile("tensor_load_to_lds %0, %1, %2, %3"
                 :: "s"(g0), "s"(g1), "s"(h2), "s"(h3)
                 : "memory");

    asm volatile("s_wait_tensorcnt 0x0" ::: "memory");

    // Per-lane reduction: lane owns H-dims [lane*8, lane*8+8).
    const float* sp = &stage[wave][0] + lane * 8;
    v4f acc0 = {0.f, 0.f, 0.f, 0.f};
    v4f acc1 = {0.f, 0.f, 0.f, 0.f};
    #pragma unroll
    for (int k = 0; k < NODES_PER_SEG; ++k) {
      acc0 += *(const v4f*)(sp + k * H_DIM);
      acc1 += *(const v4f*)(sp + k * H_DIM + 4);
    }
    acc0 *= inv_cnt;
    acc1 *= inv_cnt;

    __builtin_nontemporal_store(acc0, (v4f*)(orow + lane * 8));
    __builtin_nontemporal_store(acc1, (v4f*)(orow + lane * 8 + 4));
    t = (float)time_vals[seg];
  } else {
    // ---- pad path ----
    v4f z = {0.f, 0.f, 0.f, 0.f};
    __builtin_nontemporal_store(z, (v4f*)(orow + lane * 8));
    __builtin_nontemporal_store(z, (v4f*)(orow + lane * 8 + 4));
    t = PAD_TIME;
  }

  // Time embedding: T_DIM == 32 == lanes per wave.
  float te = cosf(t * t_w[lane] + t_b[lane]);
  __builtin_nontemporal_store(te, orow + H_DIM + lane);
}

// ---------------------------------------------------------------------------
extern "C" void kernel_launch(void* const* d_in, const int* in_sizes, int n_in,
                              void* d_out, int out_size, void* d_ws, size_t ws_size,
                              hipStream_t stream) {
  const float* ent         = (const float*)d_in[0];
  const float* t_w         = (const float*)d_in[1];
  const float* t_b         = (const float*)d_in[2];
  const int*   flat_s      = (const int*)d_in[3];
  // d_in[4] = node_seg_ids: implied by the regular NODES_PER_SEG structure.
  const int*   seg_example = (const int*)d_in[5];
  const int*   seg_pos     = (const int*)d_in[6];
  const int*   time_vals   = (const int*)d_in[7];
  float*       out         = (float*)d_out;
  int*         map         = (int*)d_ws;     // n_slots ints of scratch

  const int n_seg   = in_sizes[5];
  const int n_slots = out_size / OUT_DIM;    // B * SEQ_LEN
  const unsigned n_ent = (unsigned)(in_sizes[0] / H_DIM);
  const int nodes_per_seg = in_sizes[3] / in_sizes[5];
  const float inv_cnt = 1.0f / (float)nodes_per_seg;

  // A) map <- -1 (deterministic each call; ws is not re-poisoned by harness).
  map_init_kernel<<<(n_slots + 255) / 256, 256, 0, stream>>>(map, n_slots);

  // B) invert the scatter.
  map_scatter_kernel<<<(n_seg + 255) / 256, 256, 0, stream>>>(
      map, seg_example, seg_pos, n_seg);

  // C) one pass over the output, wave per slot, TDM-gathered embeddings.
  int blocks = (n_slots + WAVES_PER_BLOCK - 1) / WAVES_PER_BLOCK;
  out_kernel<<<blocks, WAVES_PER_BLOCK * 32, 0, stream>>>(
      ent, flat_s, time_vals, t_w, t_b, map, out, n_slots, n_ent, inv_cnt);
}